// CustomRGCN_71098888618669
// MI455X (gfx1250) — compile-verified
//
#include <hip/hip_runtime.h>

// ---------------------------------------------------------------------------
// CustomRGCN for MI455X (gfx1250).  All GEMMs run on the CDNA5 matrix pipe via
// V_WMMA_F32_16X16X4_F32 (full fp32 precision -- the workload is memory bound,
// ~43 GFLOP vs ~4 GB of traffic at 23.3 TB/s, so low-precision WMMA buys
// nothing and would diverge from the f32 reference).
//
// GEMM scheme: one workgroup per 16-row node tile; waves within the workgroup
// cover the output columns in 32-wide strips (2 WMMA accumulators per wave).
// The A tile is staged through LDS in 16x64 chunks with coalesced global
// loads; the LDS row stride is padded (+2 floats) so the row-strided fragment
// reads hit distinct banks (64 banks x 4B).  The WMMA k-loop is unrolled 4x
// so the scheduler can clause the weight loads and issue 8 WMMAs per trip.
// ---------------------------------------------------------------------------

typedef __attribute__((ext_vector_type(2))) float v2f;
typedef __attribute__((ext_vector_type(8))) float v8f;

#define DD    200   // embedding dim
#define II    40    // individual size
#define SLOPE 0.01f
#define KB    64          // K chunk staged in LDS
#define KBP   (KB + 2)    // padded LDS row stride (floats) -> conflict-free

// ---------------------------------------------------------------------------
// C[:, coff:coff+Ncols] = act(A[MxK] @ W[KxNcols] + b)
// A row-major (stride lda), W row-major (stride ldw), C row-major (stride ldc).
// grid.x = M/16 node tiles; blockDim.x = ceil(Ncols/32)*32 (one wave per
// 32-col strip).  flags: bit0 = add bias, bit1 = leaky-relu.  K % 4 == 0,
// and K % 64 is 0 or a multiple of 8 (768 and 200 both qualify).
// ---------------------------------------------------------------------------
__global__ __launch_bounds__(256)
void wmma_gemm_f32(const float* __restrict__ A, int lda,
                   const float* __restrict__ W, int ldw,
                   const float* __restrict__ bias,
                   float* __restrict__ C, int ldc, int coff,
                   int M, int K, int Ncols, int flags)
{
    __shared__ float As[16 * KBP];

    const int lane = threadIdx.x & 31;
    const int wave = threadIdx.x >> 5;
    const int half = lane >> 4;        // 0: K pair {k,k+1}, 1: K pair {k+2,k+3}
    const int l16  = lane & 15;
    const int mt   = blockIdx.x << 4;

    // This wave's two 16-col output tiles.
    const int col0  = wave * 32 + l16;
    const int col1  = col0 + 16;
    const int bcol0 = min(col0, Ncols - 1);   // clamped loads; masked stores
    const int bcol1 = min(col1, Ncols - 1);

    v8f acc0 = {};
    v8f acc1 = {};

    const float* ApL = As + l16 * KBP + 2 * half;

    for (int kb = 0; kb < K; kb += KB) {
        const int kchunk = min(KB, K - kb);

        // Cooperative, coalesced stage of A[mt:mt+16, kb:kb+64] into LDS.
#pragma unroll 4
        for (int i = threadIdx.x; i < 16 * KB; i += blockDim.x) {
            int r  = i >> 6;                   // row within tile
            int c  = i & (KB - 1);             // col within chunk
            int ar = min(mt + r, M - 1);       // clamp: garbage never consumed
            int ac = min(kb + c, K - 1);
            As[r * KBP + c] = A[(size_t)ar * lda + ac];
        }
        __syncthreads();

#pragma unroll 4
        for (int k = 0; k < kchunk; k += 4) {
            v2f a;
            a[0] = ApL[k];
            a[1] = ApL[k + 1];
            const float* Wr = W + (size_t)(kb + k + 2 * half) * ldw;
            v2f b0, b1;
            b0[0] = Wr[bcol0];
            b0[1] = Wr[ldw + bcol0];
            b1[0] = Wr[bcol1];
            b1[1] = Wr[ldw + bcol1];
            acc0 = __builtin_amdgcn_wmma_f32_16x16x4_f32(
                       false, a, false, b0, (short)0, acc0, false, false);
            acc1 = __builtin_amdgcn_wmma_f32_16x16x4_f32(
                       false, a, false, b1, (short)0, acc1, false, false);
        }
        __syncthreads();
    }

    // Epilogue (C/D layout: vgpr j -> row mt + half*8 + j, lane -> column).
    const float bv0 = (flags & 1) && (col0 < Ncols) ? bias[col0] : 0.0f;
    const float bv1 = (flags & 1) && (col1 < Ncols) ? bias[col1] : 0.0f;
#pragma unroll
    for (int j = 0; j < 8; ++j) {
        int r = mt + half * 8 + j;
        if (r < M) {
            if (col0 < Ncols) {
                float v = acc0[j] + bv0;
                if (flags & 2) v = v > 0.0f ? v : SLOPE * v;
                C[(size_t)r * ldc + coff + col0] = v;
            }
            if (col1 < Ncols) {
                float v = acc1[j] + bv1;
                if (flags & 2) v = v > 0.0f ? v : SLOPE * v;
                C[(size_t)r * ldc + coff + col1] = v;
            }
        }
    }
}

// Tiny-K encoders (num_prop K=6, cat_prop K=11): one thread per (node, out col)
__global__ void small_dense_lrelu(const float* __restrict__ in, int K,
                                  const float* __restrict__ w,  // [K x II]
                                  const float* __restrict__ b,
                                  float* __restrict__ x0, int coff, int Nn)
{
    int t = blockIdx.x * blockDim.x + threadIdx.x;
    int n = t / II, j = t % II;
    if (n >= Nn) return;
    float s = b[j];
    for (int k = 0; k < K; ++k) s += in[(size_t)n * K + k] * w[k * II + j];
    s = s > 0.0f ? s : SLOPE * s;
    x0[(size_t)n * DD + coff + j] = s;
}

// Per-(dst, relation) in-degree counts (float, so reciprocal is in place)
__global__ void count_kernel(const int* __restrict__ dst,
                             const int* __restrict__ et,
                             float* __restrict__ cnt0, float* __restrict__ cnt1, int E)
{
    int e = blockIdx.x * blockDim.x + threadIdx.x;
    if (e >= E) return;
    float* c = et[e] ? cnt1 : cnt0;
    atomicAdd(&c[dst[e]], 1.0f);
}

__global__ void recip_kernel(float* __restrict__ c, int n)
{
    int i = blockIdx.x * blockDim.x + threadIdx.x;
    if (i >= n) return;
    c[i] = 1.0f / fmaxf(c[i], 1.0f);
}

// Edge scatter: agg[etype][dst] += h[etype][src]   (one block per edge;
// reads and atomic RMWs are contiguous across the 200 channel lanes)
__global__ void scatter_kernel(const int* __restrict__ src,
                               const int* __restrict__ dst,
                               const int* __restrict__ et,
                               const float* __restrict__ h0,
                               const float* __restrict__ h1,
                               float* __restrict__ agg0,
                               float* __restrict__ agg1, int E)
{
    int e = blockIdx.x;
    int c = threadIdx.x;
    if (e >= E || c >= DD) return;
    int t = et[e];
    const float* h   = (t ? h1 : h0)     + (size_t)src[e] * DD;
    float*       agg = (t ? agg1 : agg0) + (size_t)dst[e] * DD;
    atomicAdd(&agg[c], h[c]);
}

// x (already = x@root + b) += mean_r0 + mean_r1
__global__ void combine_kernel(float* __restrict__ x,
                               const float* __restrict__ agg0,
                               const float* __restrict__ agg1,
                               const float* __restrict__ inv0,
                               const float* __restrict__ inv1, int Nn)
{
    int t = blockIdx.x * blockDim.x + threadIdx.x;
    if (t >= Nn * DD) return;
    int n = t / DD;
    x[t] += agg0[t] * inv0[n] + agg1[t] * inv1[n];
}

// Final head: [N,200] @ [200,2] + b  (too skinny for WMMA to matter)
__global__ void out3_kernel(const float* __restrict__ x,
                            const float* __restrict__ w,   // [200 x 2]
                            const float* __restrict__ b,
                            float* __restrict__ out, int Nn)
{
    int n = blockIdx.x * blockDim.x + threadIdx.x;
    if (n >= Nn) return;
    float s0 = b[0], s1 = b[1];
    const float* xr = x + (size_t)n * DD;
    for (int k = 0; k < DD; ++k) {
        float xv = xr[k];
        s0 += xv * w[2 * k + 0];
        s1 += xv * w[2 * k + 1];
    }
    out[2 * n + 0] = s0;
    out[2 * n + 1] = s1;
}

static inline void launch_gemm(const float* A, int lda, const float* W, int ldw,
                               const float* bias, float* C, int ldc, int coff,
                               int M, int K, int Nc, int flags, hipStream_t s)
{
    int mt = (M + 15) / 16;
    int nw = (Nc + 31) / 32;          // waves per block (32 cols each)
    wmma_gemm_f32<<<dim3(mt), dim3(nw * 32), 0, s>>>(A, lda, W, ldw, bias,
                                                     C, ldc, coff, M, K, Nc, flags);
}

// ---------------------------------------------------------------------------
// Input layout assumption: setup_inputs() dict flattened in insertion order,
// nested param dicts likewise (w before b inside each _lin):
//  0 num_prop 1 cat_prop 2 tweet_emb 3 user_emb 4 user_name_emb
//  5 edge_index[2,E] 6 edge_type
//  7 num.w  8 num.b  9 cat.w 10 cat.b 11 tweet.w 12 tweet.b 13 user.w 14 user.b
// 15 uname.w 16 uname.b 17 inp.w 18 inp.b 19 rgcn_w 20 rgcn_root 21 rgcn_b
// 22 out1.w 23 out1.b 24 out2.w 25 out2.b 26 out3.w 27 out3.b
// Workspace: 6 * N*200 floats + 2*N floats  (~241 MB for N=50000).
// ---------------------------------------------------------------------------
extern "C" void kernel_launch(void* const* d_in, const int* in_sizes, int n_in,
                              void* d_out, int out_size, void* d_ws, size_t ws_size,
                              hipStream_t stream)
{
    const float* num_prop  = (const float*)d_in[0];
    const float* cat_prop  = (const float*)d_in[1];
    const float* tweet_emb = (const float*)d_in[2];
    const float* user_emb  = (const float*)d_in[3];
    const float* uname_emb = (const float*)d_in[4];
    const int Nn = in_sizes[0] / 6;
    const int E  = in_sizes[6];
    const int* src = (const int*)d_in[5];
    const int* dst = src + E;
    const int* et  = (const int*)d_in[6];

    const float* w_num  = (const float*)d_in[7];  const float* b_num  = (const float*)d_in[8];
    const float* w_cat  = (const float*)d_in[9];  const float* b_cat  = (const float*)d_in[10];
    const float* w_tw   = (const float*)d_in[11]; const float* b_tw   = (const float*)d_in[12];
    const float* w_us   = (const float*)d_in[13]; const float* b_us   = (const float*)d_in[14];
    const float* w_un   = (const float*)d_in[15]; const float* b_un   = (const float*)d_in[16];
    const float* w_inp  = (const float*)d_in[17]; const float* b_inp  = (const float*)d_in[18];
    const float* rgcn_w    = (const float*)d_in[19];
    const float* rgcn_root = (const float*)d_in[20];
    const float* rgcn_b    = (const float*)d_in[21];
    const float* w_o1 = (const float*)d_in[22]; const float* b_o1 = (const float*)d_in[23];
    const float* w_o2 = (const float*)d_in[24]; const float* b_o2 = (const float*)d_in[25];
    const float* w_o3 = (const float*)d_in[26]; const float* b_o3 = (const float*)d_in[27];

    float* ws = (float*)d_ws;
    const size_t SZ = (size_t)Nn * DD;
    float* bufA = ws;            // x0 / conv1 out / out1 out
    float* bufB = ws + SZ;       // x1 / conv2 out
    float* h0   = ws + 2 * SZ;   // relation-0 transform / out2 out
    float* h1   = ws + 3 * SZ;   // relation-1 transform
    float* agg0 = ws + 4 * SZ;
    float* agg1 = ws + 5 * SZ;
    float* cnt0 = ws + 6 * SZ;   // N floats -> 1/deg after recip
    float* cnt1 = cnt0 + Nn;

    const int NB256 = 256;

    // ---- feature encoders -> x0 (bufA) --------------------------------
    {
        int tot = Nn * II;
        small_dense_lrelu<<<(tot + NB256 - 1) / NB256, NB256, 0, stream>>>(
            num_prop, 6,  w_num, b_num, bufA, 0,  Nn);
        small_dense_lrelu<<<(tot + NB256 - 1) / NB256, NB256, 0, stream>>>(
            cat_prop, 11, w_cat, b_cat, bufA, II, Nn);
    }
    launch_gemm(tweet_emb, 768, w_tw, II, b_tw, bufA, DD, 2 * II, Nn, 768, II, 3, stream);
    launch_gemm(user_emb,  768, w_us, II, b_us, bufA, DD, 3 * II, Nn, 768, II, 3, stream);
    launch_gemm(uname_emb, 768, w_un, II, b_un, bufA, DD, 4 * II, Nn, 768, II, 3, stream);

    // ---- inp dense: x (bufB) = lrelu(x0 @ Winp + b) -------------------
    launch_gemm(bufA, DD, w_inp, DD, b_inp, bufB, DD, 0, Nn, DD, DD, 3, stream);

    // ---- per-(node, relation) degrees (shared by both convs) ----------
    hipMemsetAsync(cnt0, 0, 2 * (size_t)Nn * sizeof(float), stream);
    count_kernel<<<(E + NB256 - 1) / NB256, NB256, 0, stream>>>(dst, et, cnt0, cnt1, E);
    recip_kernel<<<(2 * Nn + NB256 - 1) / NB256, NB256, 0, stream>>>(cnt0, 2 * Nn);

    const float* W0 = rgcn_w;             // [200 x 200], relation 0
    const float* W1 = rgcn_w + DD * DD;   // relation 1

    // ---- RGCN conv #1: bufB -> bufA -----------------------------------
    launch_gemm(bufB, DD, W0, DD, nullptr, h0, DD, 0, Nn, DD, DD, 0, stream);
    launch_gemm(bufB, DD, W1, DD, nullptr, h1, DD, 0, Nn, DD, DD, 0, stream);
    hipMemsetAsync(agg0, 0, 2 * SZ * sizeof(float), stream);   // agg0+agg1 contiguous
    scatter_kernel<<<E, 224, 0, stream>>>(src, dst, et, h0, h1, agg0, agg1, E);
    launch_gemm(bufB, DD, rgcn_root, DD, rgcn_b, bufA, DD, 0, Nn, DD, DD, 1, stream);
    combine_kernel<<<((int)(Nn * DD) + NB256 - 1) / NB256, NB256, 0, stream>>>(
        bufA, agg0, agg1, cnt0, cnt1, Nn);

    // ---- RGCN conv #2 (same weights): bufA -> bufB --------------------
    launch_gemm(bufA, DD, W0, DD, nullptr, h0, DD, 0, Nn, DD, DD, 0, stream);
    launch_gemm(bufA, DD, W1, DD, nullptr, h1, DD, 0, Nn, DD, DD, 0, stream);
    hipMemsetAsync(agg0, 0, 2 * SZ * sizeof(float), stream);
    scatter_kernel<<<E, 224, 0, stream>>>(src, dst, et, h0, h1, agg0, agg1, E);
    launch_gemm(bufA, DD, rgcn_root, DD, rgcn_b, bufB, DD, 0, Nn, DD, DD, 1, stream);
    combine_kernel<<<((int)(Nn * DD) + NB256 - 1) / NB256, NB256, 0, stream>>>(
        bufB, agg0, agg1, cnt0, cnt1, Nn);

    // ---- heads --------------------------------------------------------
    launch_gemm(bufB, DD, w_o1, DD, b_o1, bufA, DD, 0, Nn, DD, DD, 3, stream);
    launch_gemm(bufA, DD, w_o2, DD, b_o2, h0,   DD, 0, Nn, DD, DD, 3, stream);
    out3_kernel<<<(Nn + NB256 - 1) / NB256, NB256, 0, stream>>>(h0, w_o3, b_o3,
                                                                (float*)d_out, Nn);
}